// _ChannelAttentionModule_11407433138294
// MI455X (gfx1250) — compile-verified
//
#include <hip/hip_runtime.h>
#include <hip/hip_bf16.h>
#include <cstdint>

typedef __attribute__((ext_vector_type(2))) float v2f;
typedef __attribute__((ext_vector_type(8))) float v8f;

#define CCH   512          // channels
#define NSP   9216         // spatial (96*96)
#define NB    8            // batch
#define BM    64
#define BN    128
#define KT    32           // K tile per stage
#define AST   36           // LDS row stride for 64x32 A tile (padded, b64-aligned)
#define BST   36           // LDS row stride for 128x32 row-major B tile
#define BNS   132          // LDS row stride for 32x128 k-major B tile (kernel 3)

static __device__ __forceinline__ v8f wmma_f32(v2f a, v2f b, v8f c) {
  // V_WMMA_F32_16X16X4_F32 : D = A(16x4) * B(4x16) + C(16x16), all f32
  return __builtin_amdgcn_wmma_f32_16x16x4_f32(false, a, false, b, (short)0, c,
                                               false, false);
}

// CDNA5 async global->LDS (tracked by ASYNCcnt). INST_OFFSET is added to both
// the LDS and global addresses, so contiguous chunks share one address pair.
template <int OFF>
static __device__ __forceinline__ void async_b128(uint32_t lds, const float* g) {
  asm volatile("global_load_async_to_lds_b128 %0, %1, off offset:%2"
               :: "v"(lds), "v"((uint64_t)(uintptr_t)g), "i"(OFF)
               : "memory");
}
static __device__ __forceinline__ void wait_async_le6() {
  asm volatile("s_wait_asynccnt 0x6" ::: "memory");
}
static __device__ __forceinline__ void wait_async_0() {
  asm volatile("s_wait_asynccnt 0x0" ::: "memory");
}
// Generic LDS pointer: low 32 bits are the workgroup-relative LDS byte address.
static __device__ __forceinline__ uint32_t lds_off(const void* p) {
  return (uint32_t)(uintptr_t)p;
}

// ---------------------------------------------------------------------------
// Kernel 1: att[b] = Y[b] * Y[b]^T   (M=C, N=C, K=NSP)
// A[m][k] = y[tm+m, k]   B[k][n] = y[tn+n, k]   (both tiles row-major in LDS)
// Double-buffered LDS, async global->LDS staging one K-stage ahead.
// ---------------------------------------------------------------------------
__global__ __launch_bounds__(256) void gram_kernel(const float* __restrict__ x,
                                                   float* __restrict__ att) {
  __shared__ float As[2][BM * AST];   // 2 x (64 x 32, stride 36)
  __shared__ float Bs[2][BN * BST];   // 2 x (128 x 32, stride 36)

  const int b  = blockIdx.z;
  const int tm = blockIdx.x * BM;
  const int tn = blockIdx.y * BN;
  const float* __restrict__ y = x + (size_t)b * CCH * NSP;

  const int tid  = threadIdx.x;
  const int lane = tid & 31;
  const int wave = tid >> 5;
  const int wm   = (wave & 1) * 32;   // 2 waves over M
  const int wn   = (wave >> 1) * 32;  // 4 waves over N
  const int l15  = lane & 15;
  const int hh   = lane >> 4;

  v8f acc[2][2] = {};

  // per-thread staging assignments: A 64x32 (8 floats/thr), B 128x32 (16/thr)
  const int am = tid >> 2, aq = (tid & 3) * 8;
  const int bm = tid >> 1, bq = (tid & 1) * 16;
  const float* gA = y + (size_t)(tm + am) * NSP + aq;
  const float* gB = y + (size_t)(tn + bm) * NSP + bq;
  uint32_t lA[2], lB[2];
  lA[0] = lds_off(&As[0][am * AST + aq]);
  lA[1] = lds_off(&As[1][am * AST + aq]);
  lB[0] = lds_off(&Bs[0][bm * BST + bq]);
  lB[1] = lds_off(&Bs[1][bm * BST + bq]);

  auto issue = [&](int k0, int s) {
    const float* ga = gA + k0;
    const float* gb = gB + k0;
    async_b128<0>(lA[s], ga);
    async_b128<16>(lA[s], ga);
    async_b128<0>(lB[s], gb);
    async_b128<16>(lB[s], gb);
    async_b128<32>(lB[s], gb);
    async_b128<48>(lB[s], gb);
  };

  const int S = NSP / KT;   // 288 stages
  issue(0, 0);
  for (int s = 0; s < S; ++s) {
    if (s + 1 < S) { issue((s + 1) * KT, (s + 1) & 1); wait_async_le6(); }
    else           { wait_async_0(); }
    __syncthreads();

    const float* as = As[s & 1];
    const float* bs = Bs[s & 1];
#pragma unroll
    for (int kk = 0; kk < KT; kk += 4) {
      const int ko = kk + 2 * hh;   // lane-pair K offset per ISA A/B layout
      v2f a0 = *(const v2f*)&as[(wm + l15) * AST + ko];
      v2f a1 = *(const v2f*)&as[(wm + 16 + l15) * AST + ko];
      v2f b0 = *(const v2f*)&bs[(wn + l15) * BST + ko];
      v2f b1 = *(const v2f*)&bs[(wn + 16 + l15) * BST + ko];
      acc[0][0] = wmma_f32(a0, b0, acc[0][0]);
      acc[0][1] = wmma_f32(a0, b1, acc[0][1]);
      acc[1][0] = wmma_f32(a1, b0, acc[1][0]);
      acc[1][1] = wmma_f32(a1, b1, acc[1][1]);
    }
    __syncthreads();
  }

  float* __restrict__ ab = att + (size_t)b * CCH * CCH;
#pragma unroll
  for (int mi = 0; mi < 2; ++mi) {
#pragma unroll
    for (int ni = 0; ni < 2; ++ni) {
      const int r0 = tm + wm + mi * 16 + 8 * hh;
      const int c0 = tn + wn + ni * 16 + l15;
#pragma unroll
      for (int v = 0; v < 8; ++v)
        ab[(size_t)(r0 + v) * CCH + c0] = acc[mi][ni][v];
    }
  }
}

// ---------------------------------------------------------------------------
// Kernel 2: in-place softmax(rowmax - a) == exp(amin - a)/sum over each row
// ---------------------------------------------------------------------------
__global__ __launch_bounds__(256) void danet_softmax(float* __restrict__ att) {
  __shared__ float redm[8];
  __shared__ float reds[8];
  const int tid = threadIdx.x;
  float* a = att + (size_t)blockIdx.x * CCH;

  const float v0 = a[tid];
  const float v1 = a[tid + 256];

  float m = fminf(v0, v1);
#pragma unroll
  for (int o = 16; o > 0; o >>= 1) m = fminf(m, __shfl_down(m, o, 32));
  if ((tid & 31) == 0) redm[tid >> 5] = m;
  __syncthreads();
  if (tid < 8) {
    float t = redm[tid];
#pragma unroll
    for (int o = 4; o > 0; o >>= 1) t = fminf(t, __shfl_down(t, o, 8));
    if (tid == 0) redm[0] = t;
  }
  __syncthreads();
  const float amin = redm[0];

  const float e0 = __expf(amin - v0);
  const float e1 = __expf(amin - v1);
  float s = e0 + e1;
#pragma unroll
  for (int o = 16; o > 0; o >>= 1) s += __shfl_down(s, o, 32);
  if ((tid & 31) == 0) reds[tid >> 5] = s;
  __syncthreads();
  if (tid < 8) {
    float t = reds[tid];
#pragma unroll
    for (int o = 4; o > 0; o >>= 1) t += __shfl_down(t, o, 8);
    if (tid == 0) reds[0] = t;
  }
  __syncthreads();

  const float inv = 1.0f / reds[0];
  a[tid]       = e0 * inv;
  a[tid + 256] = e1 * inv;
}

// ---------------------------------------------------------------------------
// Kernel 3: out[b] = gamma * (att_sm[b] * Y[b]) + x[b]   (M=C, N=NSP, K=C)
// A[m][k] = att[tm+m, k] (row-major tile), B[k][n] = y[k, tn+n] (k-major tile)
// ---------------------------------------------------------------------------
__global__ __launch_bounds__(256) void feat_kernel(const float* __restrict__ att,
                                                   const float* __restrict__ x,
                                                   const float* __restrict__ gamma,
                                                   float* __restrict__ out) {
  __shared__ float As[2][BM * AST];   // 2 x (64 x 32, stride 36)
  __shared__ float Bs[2][KT * BNS];   // 2 x (32 x 128, stride 132)

  const int b  = blockIdx.z;
  const int tm = blockIdx.x * BM;
  const int tn = blockIdx.y * BN;
  const float* __restrict__ A = att + (size_t)b * CCH * CCH;
  const float* __restrict__ Y = x + (size_t)b * CCH * NSP;

  const int tid  = threadIdx.x;
  const int lane = tid & 31;
  const int wave = tid >> 5;
  const int wm   = (wave & 1) * 32;
  const int wn   = (wave >> 1) * 32;
  const int l15  = lane & 15;
  const int hh   = lane >> 4;

  v8f acc[2][2] = {};

  // A tile 64x32 (8 floats/thr), B tile 32x128 (16 floats/thr)
  const int am = tid >> 2, aq = (tid & 3) * 8;
  const int bk = tid >> 3, bq = (tid & 7) * 16;
  const float* gA = A + (size_t)(tm + am) * CCH + aq;
  const float* gB = Y + (size_t)bk * NSP + tn + bq;
  uint32_t lA[2], lB[2];
  lA[0] = lds_off(&As[0][am * AST + aq]);
  lA[1] = lds_off(&As[1][am * AST + aq]);
  lB[0] = lds_off(&Bs[0][bk * BNS + bq]);
  lB[1] = lds_off(&Bs[1][bk * BNS + bq]);

  auto issue = [&](int k0, int s) {
    const float* ga = gA + k0;
    const float* gb = gB + (size_t)k0 * NSP;
    async_b128<0>(lA[s], ga);
    async_b128<16>(lA[s], ga);
    async_b128<0>(lB[s], gb);
    async_b128<16>(lB[s], gb);
    async_b128<32>(lB[s], gb);
    async_b128<48>(lB[s], gb);
  };

  const int S = CCH / KT;   // 16 stages
  issue(0, 0);
  for (int s = 0; s < S; ++s) {
    if (s + 1 < S) { issue((s + 1) * KT, (s + 1) & 1); wait_async_le6(); }
    else           { wait_async_0(); }
    __syncthreads();

    const float* as = As[s & 1];
    const float* bs = Bs[s & 1];
#pragma unroll
    for (int kk = 0; kk < KT; kk += 4) {
      const int ko = kk + 2 * hh;
      v2f a0 = *(const v2f*)&as[(wm + l15) * AST + ko];
      v2f a1 = *(const v2f*)&as[(wm + 16 + l15) * AST + ko];
      v2f b0, b1;
      b0.x = bs[(ko)     * BNS + wn + l15];
      b0.y = bs[(ko + 1) * BNS + wn + l15];
      b1.x = bs[(ko)     * BNS + wn + 16 + l15];
      b1.y = bs[(ko + 1) * BNS + wn + 16 + l15];
      acc[0][0] = wmma_f32(a0, b0, acc[0][0]);
      acc[0][1] = wmma_f32(a0, b1, acc[0][1]);
      acc[1][0] = wmma_f32(a1, b0, acc[1][0]);
      acc[1][1] = wmma_f32(a1, b1, acc[1][1]);
    }
    __syncthreads();
  }

  const float g = gamma[0];
#pragma unroll
  for (int mi = 0; mi < 2; ++mi) {
#pragma unroll
    for (int ni = 0; ni < 2; ++ni) {
      const int r0 = tm + wm + mi * 16 + 8 * hh;
      const int c0 = tn + wn + ni * 16 + l15;
#pragma unroll
      for (int v = 0; v < 8; ++v) {
        const size_t idx = ((size_t)b * CCH + r0 + v) * NSP + c0;
        out[idx] = g * acc[mi][ni][v] + x[idx];
      }
    }
  }
}

// ---------------------------------------------------------------------------
extern "C" void kernel_launch(void* const* d_in, const int* in_sizes, int n_in,
                              void* d_out, int out_size, void* d_ws, size_t ws_size,
                              hipStream_t stream) {
  (void)in_sizes; (void)n_in; (void)out_size; (void)ws_size;
  const float* x     = (const float*)d_in[0];
  const float* gamma = (const float*)d_in[1];
  float* out         = (float*)d_out;
  float* att         = (float*)d_ws;   // NB*CCH*CCH floats = 8 MiB scratch

  dim3 blk(256, 1, 1);
  dim3 g1(CCH / BM, CCH / BN, NB);      // 8 x 4 x 8
  gram_kernel<<<g1, blk, 0, stream>>>(x, att);

  dim3 g2(NB * CCH, 1, 1);              // 4096 rows
  danet_softmax<<<g2, blk, 0, stream>>>(att);

  dim3 g3(CCH / BM, NSP / BN, NB);      // 8 x 72 x 8
  feat_kernel<<<g3, blk, 0, stream>>>(att, x, gamma, out);
}